// DGCNN_84052509982842
// MI455X (gfx1250) — compile-verified
//
#include <hip/hip_runtime.h>
#include <hip/hip_bf16.h>
#include <math.h>

// ---------------------------------------------------------------------------
// DGCNN inference for MI455X (gfx1250, wave32).
//  B=128 clouds, P=1024 points, K=4 neighbors.
//  Final MLP GEMMs run on V_WMMA_F32_16X16X4_F32 (fp32 tensor path).
//  Each wave owns a 16x272 C stripe (17 accumulators) so A is read once.
//  B panels staged to LDS with double-buffered GLOBAL_LOAD_ASYNC_TO_LDS.
// ---------------------------------------------------------------------------

#define NB 128
#define NP 1024
#define NPTS (NB * NP)   // 131072
#define KNN 4
#define NPAD 272         // 264 padded to 17 * 16
#define NT 17            // n-tiles per row stripe
#define BSTRIDE 20       // LDS column stride (pad): conflict-free b64 reads

#define USE_ASYNC_LDS 1  // flip to 0 if the async asm is rejected

typedef float v2f __attribute__((ext_vector_type(2)));
typedef float v8f __attribute__((ext_vector_type(8)));

// ---------------------------------------------------------------------------
// Zero-pad a srcRows x srcCols matrix into dstRows x dstCols.
// ---------------------------------------------------------------------------
__global__ void pad_kernel(const float* __restrict__ src, float* __restrict__ dst,
                           int srcRows, int srcCols, int dstRows, int dstCols) {
    int i = blockIdx.x * blockDim.x + threadIdx.x;
    int total = dstRows * dstCols;
    if (i >= total) return;
    int r = i / dstCols, c = i - r * dstCols;
    dst[i] = (r < srcRows && c < srcCols) ? src[r * srcCols + c] : 0.0f;
}

// ---------------------------------------------------------------------------
// Per-cloud kNN (k=4, self included). One block per cloud; cloud features
// staged in LDS (<=1024*32*4 = 128KB, fits in the 320KB WGP LDS).
// dist = |xi|^2 + |xj|^2 - 2 xi.xj  (same formula as reference).
// Stores GLOBAL row indices.
// ---------------------------------------------------------------------------
template<int D>
__global__ void knn_kernel(const float* __restrict__ X, int ldx,
                           int* __restrict__ nn) {
    extern __shared__ float sm[];
    float* xs = sm;              // NP * D
    float* sq = sm + NP * D;     // NP
    const int cloud = blockIdx.x;
    const int tid = threadIdx.x;
    const long base = (long)cloud * NP;

    for (int i = tid; i < NP * D; i += blockDim.x) {
        int p = i / D, c = i - p * D;
        xs[i] = X[(base + p) * (long)ldx + c];
    }
    __syncthreads();
    for (int p = tid; p < NP; p += blockDim.x) {
        float s = 0.0f;
        #pragma unroll
        for (int c = 0; c < D; ++c) { float v = xs[p * D + c]; s += v * v; }
        sq[p] = s;
    }
    __syncthreads();

    for (int i = tid; i < NP; i += blockDim.x) {
        float xi[D];
        #pragma unroll
        for (int c = 0; c < D; ++c) xi[c] = xs[i * D + c];
        const float si = sq[i];
        float bd0 = 3.0e38f, bd1 = 3.0e38f, bd2 = 3.0e38f, bd3 = 3.0e38f;
        int   bi0 = 0, bi1 = 0, bi2 = 0, bi3 = 0;
        for (int j = 0; j < NP; ++j) {
            float dot = 0.0f;
            #pragma unroll
            for (int c = 0; c < D; ++c) dot = fmaf(xi[c], xs[j * D + c], dot);
            float dist = si + sq[j] - 2.0f * dot;
            if (dist < bd3) {                       // strict < keeps earliest index on ties
                if (dist < bd0) {
                    bd3 = bd2; bi3 = bi2; bd2 = bd1; bi2 = bi1;
                    bd1 = bd0; bi1 = bi0; bd0 = dist; bi0 = j;
                } else if (dist < bd1) {
                    bd3 = bd2; bi3 = bi2; bd2 = bd1; bi2 = bi1;
                    bd1 = dist; bi1 = j;
                } else if (dist < bd2) {
                    bd3 = bd2; bi3 = bi2; bd2 = dist; bi2 = j;
                } else {
                    bd3 = dist; bi3 = j;
                }
            }
        }
        int* o = nn + (base + i) * KNN;
        o[0] = (int)base + bi0; o[1] = (int)base + bi1;
        o[2] = (int)base + bi2; o[3] = (int)base + bi3;
    }
}

// ---------------------------------------------------------------------------
// EdgeConv: e = [x_i, x_j - x_i] (2D) -> relu(e@W1+b1) (C) -> @W2+b2 (C),
// max over the 4 neighbors. C threads per point, weights cached in LDS.
// ---------------------------------------------------------------------------
template<int D, int C>
__global__ void edgeconv_kernel(const float* __restrict__ X, int ldx,
                                const int* __restrict__ nn,
                                const float* __restrict__ W1, const float* __restrict__ b1,
                                const float* __restrict__ W2, const float* __restrict__ b2,
                                float* __restrict__ out, int ldo) {
    constexpr int PPB = 256 / C;          // points per block
    __shared__ float W1s[2 * D * C];
    __shared__ float W2s[C * C];
    __shared__ float xps[PPB][D];
    __shared__ float xjs[PPB][D];
    __shared__ float hid[PPB][C];

    const int tid = threadIdx.x;
    const int g = tid / C;
    const int t = tid - g * C;
    const long p = (long)blockIdx.x * PPB + g;

    for (int i = tid; i < 2 * D * C; i += 256) W1s[i] = W1[i];
    for (int i = tid; i < C * C; i += 256)     W2s[i] = W2[i];
    if (t < D) xps[g][t] = X[p * (long)ldx + t];
    __syncthreads();

    float xp[D];
    #pragma unroll
    for (int c = 0; c < D; ++c) xp[c] = xps[g][c];
    const float bb1 = b1[t], bb2 = b2[t];
    float acc = -3.0e38f;

    for (int e = 0; e < KNN; ++e) {
        long j = nn[p * KNN + e];
        if (t < D) xjs[g][t] = X[j * (long)ldx + t];
        __syncthreads();
        float h = bb1;
        #pragma unroll
        for (int c = 0; c < D; ++c) {
            h = fmaf(xp[c], W1s[c * C + t], h);
            h = fmaf(xjs[g][c] - xp[c], W1s[(D + c) * C + t], h);
        }
        h = fmaxf(h, 0.0f);
        hid[g][t] = h;
        __syncthreads();
        float o = bb2;
        for (int s = 0; s < C; ++s) o = fmaf(hid[g][s], W2s[s * C + t], o);
        acc = fmaxf(acc, o);
    }
    out[p * (long)ldo + t] = acc;
}

// ---------------------------------------------------------------------------
// Async staging of one 16 x NPAD B chunk into LDS, transposed to
// buf[c * BSTRIDE + r]. 17 GLOBAL_LOAD_ASYNC_TO_LDS_B32 per thread
// (4352 dwords / 256 threads) -> ASYNCcnt += 17 per wave.
// ---------------------------------------------------------------------------
#if USE_ASYNC_LDS
__device__ __forceinline__ void stage_async(const float* __restrict__ Bm, int kc,
                                            float* buf, int tid) {
    #pragma unroll
    for (int q = 0; q < 17; ++q) {
        int i = tid + q * 256;
        int r = i / NPAD, c = i - r * NPAD;            // consecutive tid -> consecutive c
        unsigned ldst = (unsigned)(unsigned long long)(const void*)(buf + c * BSTRIDE + r);
        const float* g = Bm + (long)(kc + r) * NPAD + c;
        asm volatile("global_load_async_to_lds_b32 %0, %1, off"
                     :: "v"(ldst), "v"(g) : "memory");
    }
}
#endif

// ---------------------------------------------------------------------------
// fp32 GEMM via V_WMMA_F32_16X16X4_F32.
//   C[M x NPAD] = relu?(A[M x K] @ Bm[K x NPAD] + bias)
// Block = 256 threads (8 waves); blockIdx.x -> 128-row stripe (16 per wave).
// Each wave keeps ALL 17 n-tile accumulators (136 VGPRs) so A is streamed
// exactly once; B is staged in transposed 16xNPAD LDS chunks (stride-20
// columns => each B fragment is one aligned, bank-conflict-free ds_load_b64),
// double-buffered with async loads overlapped against the WMMA chain.
// A fragment layout (ISA 16x4 f32): lanes 0-15 hold M=lane, K=k..k+1;
// lanes 16-31 hold M=lane-16, K=k+2..k+3. C layout: VGPR r -> rows r / r+8.
// ---------------------------------------------------------------------------
__global__ void __launch_bounds__(256)
gemm_wmma_kernel(const float* __restrict__ A, int lda,
                 const float* __restrict__ Bm,
                 const float* __restrict__ bias,
                 float* __restrict__ Cm, int ldc,
                 int K, int relu) {
    __shared__ float Bs[2][NPAD * BSTRIDE];
    __shared__ float bs[NPAD];
    const int tid = threadIdx.x;

    for (int i = tid; i < NPAD; i += 256) bs[i] = bias[i];

    const int wave = tid >> 5;
    const int lane = tid & 31;
    const int ln = lane & 15;
    const int m16 = (blockIdx.x * 8 + wave) * 16;
    const int koff = (lane >> 4) * 2;
    const float* arow = A + (long)(m16 + ln) * lda + koff;

    v8f acc[NT];
    #pragma unroll
    for (int t = 0; t < NT; ++t) acc[t] = (v8f){};

    const int nchunk = K >> 4;

#if USE_ASYNC_LDS
    stage_async(Bm, 0, Bs[0], tid);
    for (int ci = 0; ci < nchunk; ++ci) {
        float* cur = Bs[ci & 1];
        if (ci + 1 < nchunk) {
            stage_async(Bm, (ci + 1) << 4, Bs[(ci + 1) & 1], tid);
            asm volatile("s_wait_asynccnt 0x11" ::: "memory");   // drain chunk ci (17 left in flight)
        } else {
            asm volatile("s_wait_asynccnt 0x0" ::: "memory");
        }
        __syncthreads();
        const int kc = ci << 4;
        __builtin_prefetch(arow + kc + 16, 0, 1);
        #pragma unroll
        for (int k0 = 0; k0 < 16; k0 += 4) {
            v2f a = *(const v2f*)(arow + kc + k0);
            v2f bf[NT];
            #pragma unroll
            for (int t = 0; t < NT; ++t)
                bf[t] = *(const v2f*)(cur + (t * 16 + ln) * BSTRIDE + k0 + koff);
            #pragma unroll
            for (int t = 0; t < NT; ++t)
                acc[t] = __builtin_amdgcn_wmma_f32_16x16x4_f32(
                    false, a, false, bf[t], (short)0, acc[t], false, false);
        }
        __syncthreads();
    }
#else
    for (int ci = 0; ci < nchunk; ++ci) {
        const int kc = ci << 4;
        __syncthreads();
        for (int q = 0; q < 17; ++q) {
            int i = tid + q * 256;
            int r = i / NPAD, c = i - r * NPAD;
            Bs[0][c * BSTRIDE + r] = Bm[(long)(kc + r) * NPAD + c];
        }
        __syncthreads();
        __builtin_prefetch(arow + kc + 16, 0, 1);
        #pragma unroll
        for (int k0 = 0; k0 < 16; k0 += 4) {
            v2f a = *(const v2f*)(arow + kc + k0);
            v2f bf[NT];
            #pragma unroll
            for (int t = 0; t < NT; ++t)
                bf[t] = *(const v2f*)(&Bs[0][0] + (t * 16 + ln) * BSTRIDE + k0 + koff);
            #pragma unroll
            for (int t = 0; t < NT; ++t)
                acc[t] = __builtin_amdgcn_wmma_f32_16x16x4_f32(
                    false, a, false, bf[t], (short)0, acc[t], false, false);
        }
    }
#endif

    const int rbase = m16 + ((lane >> 4) ? 8 : 0);
    #pragma unroll
    for (int t = 0; t < NT; ++t) {
        const int colg = t * 16 + ln;
        const float bv = bs[colg];
        #pragma unroll
        for (int r = 0; r < 8; ++r) {
            float v = acc[t][r] + bv;
            if (relu) v = fmaxf(v, 0.0f);
            Cm[(long)(rbase + r) * ldc + colg] = v;
        }
    }
}

// ---------------------------------------------------------------------------
// Final 264->2 linear + log_softmax. One wave per point.
// ---------------------------------------------------------------------------
__global__ void final_kernel(const float* __restrict__ H,
                             const float* __restrict__ W4,
                             const float* __restrict__ b4,
                             float* __restrict__ out) {
    const int lane = threadIdx.x & 31;
    const long p = (long)blockIdx.x * 8 + (threadIdx.x >> 5);
    const float* h = H + p * NPAD;
    float s0 = 0.0f, s1 = 0.0f;
    for (int k = lane; k < 264; k += 32) {
        float a = h[k];
        s0 = fmaf(a, W4[k * 2 + 0], s0);
        s1 = fmaf(a, W4[k * 2 + 1], s1);
    }
    #pragma unroll
    for (int m = 16; m >= 1; m >>= 1) {
        s0 += __shfl_xor(s0, m, 32);
        s1 += __shfl_xor(s1, m, 32);
    }
    if (lane == 0) {
        float l0 = s0 + b4[0], l1 = s1 + b4[1];
        float mx = fmaxf(l0, l1);
        float lse = mx + logf(expf(l0 - mx) + expf(l1 - mx));
        out[p * 2 + 0] = l0 - lse;
        out[p * 2 + 1] = l1 - lse;
    }
}

// ---------------------------------------------------------------------------
static inline size_t alignup(size_t x) { return (x + 255) & ~(size_t)255; }

extern "C" void kernel_launch(void* const* d_in, const int* in_sizes, int n_in,
                              void* d_out, int out_size, void* d_ws, size_t ws_size,
                              hipStream_t stream) {
    const float* x    = (const float*)d_in[0];
    const float* c1W1 = (const float*)d_in[2];
    const float* c1b1 = (const float*)d_in[3];
    const float* c1W2 = (const float*)d_in[4];
    const float* c1b2 = (const float*)d_in[5];
    const float* c2W1 = (const float*)d_in[6];
    const float* c2b1 = (const float*)d_in[7];
    const float* c2W2 = (const float*)d_in[8];
    const float* c2b2 = (const float*)d_in[9];
    const float* c3W1 = (const float*)d_in[10];
    const float* c3b1 = (const float*)d_in[11];
    const float* c3W2 = (const float*)d_in[12];
    const float* c3b2 = (const float*)d_in[13];
    const float* mW1  = (const float*)d_in[14];
    const float* mb1  = (const float*)d_in[15];
    const float* mW2  = (const float*)d_in[16];
    const float* mb2  = (const float*)d_in[17];
    const float* mW3  = (const float*)d_in[18];
    const float* mb3  = (const float*)d_in[19];
    const float* mW4  = (const float*)d_in[20];
    const float* mb4  = (const float*)d_in[21];
    float* out = (float*)d_out;

    // Workspace layout (~355 MB total).
    char* ws = (char*)d_ws;
    size_t cur = 0;
    int*   nn  = (int*)(ws + cur);   cur = alignup(cur + (size_t)NPTS * KNN * 4);
    float* f   = (float*)(ws + cur); cur = alignup(cur + (size_t)NPTS * 128 * 4);
    float* hA  = (float*)(ws + cur); cur = alignup(cur + (size_t)NPTS * NPAD * 4);
    float* hB  = (float*)(ws + cur); cur = alignup(cur + (size_t)NPTS * NPAD * 4);
    float* Wp1 = (float*)(ws + cur); cur = alignup(cur + (size_t)128 * NPAD * 4);
    float* Wp2 = (float*)(ws + cur); cur = alignup(cur + (size_t)NPAD * NPAD * 4);
    float* Wp3 = (float*)(ws + cur); cur = alignup(cur + (size_t)NPAD * NPAD * 4);
    float* bp1 = (float*)(ws + cur); cur = alignup(cur + (size_t)NPAD * 4);
    float* bp2 = (float*)(ws + cur); cur = alignup(cur + (size_t)NPAD * 4);
    float* bp3 = (float*)(ws + cur); cur = alignup(cur + (size_t)NPAD * 4);

    // Zero-pad MLP weights/biases to multiples of 16 columns/rows.
    {
        int n1 = 128 * NPAD, n2 = NPAD * NPAD, nb = NPAD;
        pad_kernel<<<(n1 + 255) / 256, 256, 0, stream>>>(mW1, Wp1, 128, 264, 128, NPAD);
        pad_kernel<<<(n2 + 255) / 256, 256, 0, stream>>>(mW2, Wp2, 264, 264, NPAD, NPAD);
        pad_kernel<<<(n2 + 255) / 256, 256, 0, stream>>>(mW3, Wp3, 264, 264, NPAD, NPAD);
        pad_kernel<<<(nb + 255) / 256, 256, 0, stream>>>(mb1, bp1, 1, 264, 1, NPAD);
        pad_kernel<<<(nb + 255) / 256, 256, 0, stream>>>(mb2, bp2, 1, 264, 1, NPAD);
        pad_kernel<<<(nb + 255) / 256, 256, 0, stream>>>(mb3, bp3, 1, 264, 1, NPAD);
    }

    // f columns: x1 -> [0,32), x2 -> [32,64), x3 -> [64,128)
    float* x1 = f + 0;
    float* x2 = f + 32;
    float* x3 = f + 64;

    // conv1: kNN on raw x (d=1), edge MLP [2->32->32]
    knn_kernel<1><<<NB, 256, (NP * 1 + NP) * 4, stream>>>(x, 1, nn);
    edgeconv_kernel<1, 32><<<NPTS / 8, 256, 0, stream>>>(
        x, 1, nn, c1W1, c1b1, c1W2, c1b2, x1, 128);

    // conv2: kNN on x1 (d=32), edge MLP [64->32->32]
    knn_kernel<32><<<NB, 256, (NP * 32 + NP) * 4, stream>>>(x1, 128, nn);
    edgeconv_kernel<32, 32><<<NPTS / 8, 256, 0, stream>>>(
        x1, 128, nn, c2W1, c2b1, c2W2, c2b2, x2, 128);

    // conv3: kNN on x2 (d=32), edge MLP [64->64->64]
    knn_kernel<32><<<NB, 256, (NP * 32 + NP) * 4, stream>>>(x2, 128, nn);
    edgeconv_kernel<32, 64><<<NPTS / 4, 256, 0, stream>>>(
        x2, 128, nn, c3W1, c3b1, c3W2, c3b2, x3, 128);

    // Final MLP: three WMMA GEMMs (fp32 tensor path) + tiny last layer.
    dim3 gblk(256);
    dim3 g1(NPTS / 128);
    gemm_wmma_kernel<<<g1, gblk, 0, stream>>>(f,  128,  Wp1, bp1, hA, NPAD, 128,  1);
    gemm_wmma_kernel<<<g1, gblk, 0, stream>>>(hA, NPAD, Wp2, bp2, hB, NPAD, NPAD, 1);
    gemm_wmma_kernel<<<g1, gblk, 0, stream>>>(hB, NPAD, Wp3, bp3, hA, NPAD, NPAD, 1);

    final_kernel<<<NPTS / 8, 256, 0, stream>>>(hA, mW4, mb4, out);

    (void)in_sizes; (void)n_in; (void)out_size; (void)ws_size;
}